// Jacobianloss_21053929685245
// MI455X (gfx1250) — compile-verified
//
#include <hip/hip_runtime.h>
#include <stdint.h>

// ---------------------------------------------------------------------------
// Jacobian log-det^2 loss: x (B,C,D,H,W) fp32 -> out (B,D,H,W) fp32
// HBM-bound stencil; TDM (tensor_load_to_lds) stages halo tiles into LDS.
// D tiled by 2 to cut plane-staging redundancy from 3x to 2x.
// ---------------------------------------------------------------------------

#define B_  2
#define C_  3
#define D_  128
#define H_  192
#define W_  192
#define HT  8            // H rows computed per block
#define RT  (HT + 2)     // staged rows incl. halo
#define NHT (H_ / HT)    // 24 strips
#define DT  2            // D planes computed per block
#define KP  (DT + 2)     // staged planes incl. halo
#define ND  (D_ / DT)    // 64 plane-pairs

typedef unsigned int u32x4 __attribute__((ext_vector_type(4)));
typedef int          i32x4 __attribute__((ext_vector_type(4)));
typedef int          i32x8 __attribute__((ext_vector_type(8)));

// Issue one TDM 2D-tile load: nrows x W_ fp32 from global -> LDS (contiguous).
// Descriptor per CDNA5 ISA (D# group0/group1 bitfields).
__device__ __forceinline__ void tdm_load_2d(unsigned lds_off, uint64_t gaddr, int nrows) {
  // force descriptor words into SGPRs (values are block-uniform)
  unsigned w1 = (unsigned)__builtin_amdgcn_readfirstlane((int)lds_off);
  unsigned w2 = (unsigned)__builtin_amdgcn_readfirstlane((int)(unsigned)(gaddr & 0xFFFFFFFFull));
  unsigned w3 = (unsigned)__builtin_amdgcn_readfirstlane(
      (int)((((unsigned)(gaddr >> 32)) & 0x01FFFFFFu) | (2u << 30)));  // addr[56:32] | type=2
  int nr = __builtin_amdgcn_readfirstlane(nrows);

  u32x4 g0;
  g0[0] = 1u;   // count=1, is_restore=0, gather off
  g0[1] = w1;   // lds_addr (bytes)
  g0[2] = w2;   // global_addr[31:0]
  g0[3] = w3;   // global_addr[56:32], type=2

  i32x8 g1;
  g1[0] = (int)(2u << 16);                          // wg_mask=0, data_size=4B
  g1[1] = (int)((unsigned)W_ << 16);                // tensor_dim0 lo16 = W
  g1[2] = (int)(((unsigned)nr & 0xFFFFu) << 16);    // dim0 hi=0 | tensor_dim1 lo16 = nrows
  g1[3] = (int)((unsigned)W_ << 16);                // dim1 hi=0 | tile_dim0 = W
  g1[4] = nr;                                       // tile_dim1 = nrows, tile_dim2 = 0
  g1[5] = W_;                                       // tensor_dim0_stride lo32 = W
  g1[6] = 0;                                        // stride hi | dim1_stride lo (unused, 2D)
  g1[7] = 0;

  i32x4 z4 = {0, 0, 0, 0};
#if __clang_major__ >= 23
  i32x8 z8 = {0, 0, 0, 0, 0, 0, 0, 0};
  __builtin_amdgcn_tensor_load_to_lds(g0, g1, z4, z4, z8, 0);
#else
  __builtin_amdgcn_tensor_load_to_lds(g0, g1, z4, z4, 0);
#endif
}

__global__ void __launch_bounds__(256)
jac_logdet_sq_kernel(const float* __restrict__ x, float* __restrict__ out) {
  // [channel][d-plane: d0-1..d0+2][row][w] = 3*4*10*192*4 = 92,160 B
  __shared__ float tile[C_][KP][RT][W_];

  const int bid = blockIdx.x;
  const int ht  = bid % NHT;
  const int dt  = (bid / NHT) % ND;
  const int b   = bid / (NHT * ND);
  const int h0  = ht * HT;
  const int d0  = dt * DT;

  // Wave 0 issues all 12 TDM tile loads (scalar branch via readfirstlane;
  // TDM ignores EXEC, one issue per instruction per wave).
  if (__builtin_amdgcn_readfirstlane((int)threadIdx.x) < 32) {
    const int lo    = (h0 > 0) ? (h0 - 1) : 0;                 // first staged row
    const int hi    = (h0 + HT < H_) ? (h0 + HT) : (H_ - 1);   // last staged row
    const int nrows = hi - lo + 1;                             // 9 or 10
    const int rofs  = lo - (h0 - 1);                           // LDS row of 'lo' (0 or 1)
#pragma unroll
    for (int c = 0; c < C_; ++c) {
#pragma unroll
      for (int k = 0; k < KP; ++k) {
        int dk = d0 + k - 1;
        dk = dk < 0 ? 0 : (dk > D_ - 1 ? D_ - 1 : dk);  // clamped plane (unused at bounds)
        const float* gp =
            x + ((((size_t)(b * C_ + c) * D_) + dk) * H_ + lo) * W_;
        unsigned lds_off = (unsigned)(uintptr_t)(&tile[c][k][rofs][0]);
        tdm_load_2d(lds_off, (uint64_t)(uintptr_t)gp, nrows);
      }
    }
    __builtin_amdgcn_s_wait_tensorcnt(0);   // all 12 DMAs landed in LDS
  }
  __syncthreads();

  const int t = threadIdx.x;
#pragma unroll
  for (int it = 0; it < (DT * HT * W_) / 256; ++it) {  // 12 voxels/thread
    const int idx = t + it * 256;
    const int dd  = idx / (HT * W_);        // 0..DT-1
    const int rem = idx - dd * (HT * W_);
    const int hr  = rem / W_;
    const int w   = rem - hr * W_;
    const int h   = h0 + hr;
    const int d   = d0 + dd;
    const int r   = hr + 1;                 // LDS row of h (row 0 = h0-1 halo)
    const int kc  = dd + 1;                 // LDS plane of d (plane 0 = d0-1 halo)

    float J[3][3];
#pragma unroll
    for (int c = 0; c < C_; ++c) {
      const float ctr = tile[c][kc][r][w];
      float gW, gH, gD;
      // W axis: central interior, one-sided at bounds (numpy gradient)
      if (w == 0)            gW = tile[c][kc][r][1] - ctr;
      else if (w == W_ - 1)  gW = ctr - tile[c][kc][r][W_ - 2];
      else                   gW = 0.5f * (tile[c][kc][r][w + 1] - tile[c][kc][r][w - 1]);
      // H axis
      if (h == 0)            gH = tile[c][kc][r + 1][w] - ctr;
      else if (h == H_ - 1)  gH = ctr - tile[c][kc][r - 1][w];
      else                   gH = 0.5f * (tile[c][kc][r + 1][w] - tile[c][kc][r - 1][w]);
      // D axis
      if (d == 0)            gD = tile[c][kc + 1][r][w] - ctr;
      else if (d == D_ - 1)  gD = ctr - tile[c][kc - 1][r][w];
      else                   gD = 0.5f * (tile[c][kc + 1][r][w] - tile[c][kc - 1][r][w]);

      J[0][c] = gD;  // row j = derivative axis (D,H,W); col i = channel
      J[1][c] = gH;
      J[2][c] = gW;
    }
    J[0][0] += 1.0f; J[1][1] += 1.0f; J[2][2] += 1.0f;

    const float det = J[0][0] * (J[1][1] * J[2][2] - J[1][2] * J[2][1])
                    - J[0][1] * (J[1][0] * J[2][2] - J[1][2] * J[2][0])
                    + J[0][2] * (J[1][0] * J[2][1] - J[1][1] * J[2][0]);
    const float l = logf(det);
    out[(((size_t)(b * D_ + d) * H_) + h) * W_ + w] = l * l;
  }
}

extern "C" void kernel_launch(void* const* d_in, const int* in_sizes, int n_in,
                              void* d_out, int out_size, void* d_ws, size_t ws_size,
                              hipStream_t stream) {
  (void)in_sizes; (void)n_in; (void)d_ws; (void)ws_size; (void)out_size;
  const float* x = (const float*)d_in[0];
  float* out = (float*)d_out;
  dim3 grid(B_ * ND * NHT);   // 3072 blocks
  dim3 block(256);            // 8 wave32
  jac_logdet_sq_kernel<<<grid, block, 0, stream>>>(x, out);
}